// PhysicsLoss_23244363006241
// MI455X (gfx1250) — compile-verified
//
#include <hip/hip_runtime.h>
#include <math.h>
#include <stdint.h>

#define N_FRAMES 256
#define DEPTH 4  // async pipeline depth (power of two)

// ---------------------------------------------------------------------------
// CDNA5 async global->LDS copy (ASYNCcnt-tracked), inline asm per
// CDNA5 ISA: per-lane 16B copy, VDST = LDS byte address,
// VADDR = 64-bit global address. Non-temporal: the heatmaps are a one-pass
// ~197MB stream (~= 192MB L2), so do not install at regular priority.
// ---------------------------------------------------------------------------
__device__ __forceinline__ void async_load_b128_nt(uint32_t lds_addr,
                                                   uint64_t gaddr) {
  asm volatile("global_load_async_to_lds_b128 %0, %1, off th:TH_LOAD_NT"
               :
               : "v"(lds_addr), "v"(gaddr)
               : "memory");
}

template <int CNT>
__device__ __forceinline__ void wait_asynccnt() {
  asm volatile("s_wait_asynccnt %0" : : "i"(CNT) : "memory");
}

// ---------------------------------------------------------------------------
// Kernel 1: row argmax. One wave32 per row. DEPTH-deep async pipeline of
// 512B LDS chunks: issue chunk c+3, wait until <=3 outstanding (chunk c
// complete, async loads complete in order), reduce chunk c from LDS.
// First-occurrence tie-break to match jnp.argmax.
// ---------------------------------------------------------------------------
__global__ void __launch_bounds__(32)
row_argmax_kernel(const float* __restrict__ src, float* __restrict__ coords,
                  int rowLen, int comp) {
  const int row = blockIdx.x;
  const int lane = threadIdx.x;  // 0..31, wave32
  const uint32_t rowBytes = (uint32_t)rowLen * 4u;
  const uint64_t rowBase =
      (uint64_t)(uintptr_t)src + (uint64_t)row * (uint64_t)rowBytes;

  __shared__ float lbuf[DEPTH][128];  // DEPTH x 512B buffers (32 lanes x 16B)
  const uint32_t ldsBase = (uint32_t)(uintptr_t)(&lbuf[0][0]);

  const int nch = (int)((rowBytes + 511u) / 512u);
  const uint32_t lim = rowBytes - 16u;  // rows are multiples of 16B

  auto chunkOff = [&](int c) -> uint32_t {
    uint32_t off = (uint32_t)c * 512u + (uint32_t)lane * 16u;
    return off > lim ? lim : off;  // clamped lanes re-read valid tail (idempotent)
  };
  auto issue = [&](int c) {
    async_load_b128_nt(
        ldsBase + (uint32_t)(c & (DEPTH - 1)) * 512u + (uint32_t)lane * 16u,
        rowBase + chunkOff(c));
  };

  float bestV = -INFINITY;
  int bestI = 0;

  auto process = [&](int c) {
    const int cur = c & (DEPTH - 1);
    const uint32_t off = chunkOff(c);
    const int idx0 = (int)(off >> 2);
    const float4 q =
        *reinterpret_cast<const float4*>(&lbuf[cur][lane * 4]);  // ds_load_b128
    const float vv[4] = {q.x, q.y, q.z, q.w};
#pragma unroll
    for (int j = 0; j < 4; ++j) {
      const float v = vv[j];
      const int idx = idx0 + j;
      if (v > bestV || (v == bestV && idx < bestI)) {
        bestV = v;
        bestI = idx;
      }
    }
  };

  // prologue: put up to 3 chunks in flight
  issue(0);
  if (nch > 1) issue(1);
  if (nch > 2) issue(2);

  int c = 0;
  // steady state: 4 outstanding after each issue; wait <=3 -> chunk c done
  for (; c + 3 < nch; ++c) {
    issue(c + 3);
    wait_asynccnt<3>();
    process(c);
  }
  // epilogue: drain remaining (1..3) outstanding chunks
  int rem = nch - c;
  if (rem == 3) {
    wait_asynccnt<2>();
    process(c++);
    rem = 2;
  }
  if (rem == 2) {
    wait_asynccnt<1>();
    process(c++);
    rem = 1;
  }
  if (rem == 1) {
    wait_asynccnt<0>();
    process(c++);
  }

  // wave32 (value, index) max-reduction, lower index wins ties
#pragma unroll
  for (int d = 16; d > 0; d >>= 1) {
    float ov = __shfl_xor(bestV, d, 32);
    int oi = __shfl_xor(bestI, d, 32);
    if (ov > bestV || (ov == bestV && oi < bestI)) {
      bestV = ov;
      bestI = oi;
    }
  }
  if (lane == 0) coords[(size_t)row * 2 + comp] = (float)bestI;
}

// ---------------------------------------------------------------------------
// Block-wide sum over blockDim.x (power of two) threads via LDS tree.
// ---------------------------------------------------------------------------
__device__ __forceinline__ float block_sum(float v, float* s_red, int tid,
                                           int nthr) {
  s_red[tid] = v;
  __syncthreads();
  for (int s = nthr >> 1; s > 0; s >>= 1) {
    if (tid < s) s_red[tid] += s_red[tid + s];
    __syncthreads();
  }
  float r = s_red[0];
  __syncthreads();
  return r;
}

// ---------------------------------------------------------------------------
// Kernel 2: one block (256 threads = 8 wave32) per batch. Stable compaction
// of valid frames, dt from frame-index gaps, vel/acc for predicted coords and
// labels, masked-loss partial sums written per block (deterministic; no
// atomics). partials[b] = {vel_num, vel_den, acc_num, acc_den}
// ---------------------------------------------------------------------------
__global__ void __launch_bounds__(N_FRAMES)
physics_loss_kernel(const float* __restrict__ labels,
                    const float* __restrict__ coords,
                    float* __restrict__ partials, int N) {
  const int b = blockIdx.x;
  const int n = threadIdx.x;

  __shared__ float s_lab[N_FRAMES][2];
  __shared__ int s_valid[N_FRAMES];
  __shared__ int s_order[N_FRAMES];
  __shared__ float s_cp[N_FRAMES][2];
  __shared__ float s_ca[N_FRAMES][2];
  __shared__ float s_vp[N_FRAMES][2];
  __shared__ float s_va[N_FRAMES][2];
  __shared__ float s_dt[N_FRAMES];
  __shared__ int s_K;
  __shared__ float s_red[N_FRAMES];

  const size_t base = (size_t)b * N + n;
  const float l0 = labels[base * 2 + 0];
  const float l1 = labels[base * 2 + 1];
  const int v = (l0 != 0.0f) && (l1 != 0.0f);
  s_lab[n][0] = l0;
  s_lab[n][1] = l1;
  s_valid[n] = v;
  __syncthreads();

  // stable compaction: valid frames first (in order), then invalid (in order)
  if (n == 0) {
    int pos = 0;
    for (int i = 0; i < N; ++i)
      if (s_valid[i]) s_order[pos++] = i;
    s_K = pos;
    for (int i = 0; i < N; ++i)
      if (!s_valid[i]) s_order[pos++] = i;
  }
  __syncthreads();

  const int srcI = s_order[n];
  const size_t sb = (size_t)b * N + srcI;
  s_cp[n][0] = coords[sb * 2 + 0];
  s_cp[n][1] = coords[sb * 2 + 1];
  s_ca[n][0] = s_lab[srcI][0];
  s_ca[n][1] = s_lab[srcI][1];
  __syncthreads();

  const int K = s_K;
  if (n < N - 1) {
    float dt = (float)(s_order[n + 1] - s_order[n]) / 25.0f;  // FPS
    s_dt[n] = dt;
    float sdt = (dt == 0.0f) ? 1.0f : dt;
    bool vm = (n < K - 1);
    s_vp[n][0] = vm ? (s_cp[n + 1][0] - s_cp[n][0]) / sdt : 0.0f;
    s_vp[n][1] = vm ? (s_cp[n + 1][1] - s_cp[n][1]) / sdt : 0.0f;
    s_va[n][0] = vm ? (s_ca[n + 1][0] - s_ca[n][0]) / sdt : 0.0f;
    s_va[n][1] = vm ? (s_ca[n + 1][1] - s_ca[n][1]) / sdt : 0.0f;
  }
  __syncthreads();

  float vnum = 0.0f, vden = 0.0f, anum = 0.0f, aden = 0.0f;
  if (n < N - 1) {
    // NOTE: loss mask is the ORIGINAL-position valid[:, 1:] (as in reference)
    float m = s_valid[n + 1] ? 1.0f : 0.0f;
    float dx = s_vp[n][0] - s_va[n][0];
    float dy = s_vp[n][1] - s_va[n][1];
    vnum = (dx * dx + dy * dy) * m;
    vden = m;
  }
  if (n < N - 2) {
    float adt = s_dt[n + 1];
    float sadt = (adt == 0.0f) ? 1.0f : adt;
    bool am = (n < K - 2);
    float apx = am ? (s_vp[n + 1][0] - s_vp[n][0]) / sadt : 0.0f;
    float apy = am ? (s_vp[n + 1][1] - s_vp[n][1]) / sadt : 0.0f;
    float aax = am ? (s_va[n + 1][0] - s_va[n][0]) / sadt : 0.0f;
    float aay = am ? (s_va[n + 1][1] - s_va[n][1]) / sadt : 0.0f;
    float m = s_valid[n + 2] ? 1.0f : 0.0f;
    float dx = apx - aax;
    float dy = apy - aay;
    anum = (dx * dx + dy * dy) * m;
    aden = m;
  }

  vnum = block_sum(vnum, s_red, n, N);
  vden = block_sum(vden, s_red, n, N);
  anum = block_sum(anum, s_red, n, N);
  aden = block_sum(aden, s_red, n, N);
  if (n == 0) {
    partials[b * 4 + 0] = vnum;
    partials[b * 4 + 1] = vden;
    partials[b * 4 + 2] = anum;
    partials[b * 4 + 3] = aden;
  }
}

// ---------------------------------------------------------------------------
// Kernel 3: deterministic final combine.
// ---------------------------------------------------------------------------
__global__ void finalize_kernel(const float* __restrict__ partials,
                                float* __restrict__ out, int B) {
  if (blockIdx.x == 0 && threadIdx.x == 0) {
    float vn = 0.0f, vd = 0.0f, an = 0.0f, ad = 0.0f;
    for (int b = 0; b < B; ++b) {
      vn += partials[b * 4 + 0];
      vd += partials[b * 4 + 1];
      an += partials[b * 4 + 2];
      ad += partials[b * 4 + 3];
    }
    out[0] = 1e-7f * (vn / vd + 0.1f * (an / ad));
  }
}

extern "C" void kernel_launch(void* const* d_in, const int* in_sizes, int n_in,
                              void* d_out, int out_size, void* d_ws,
                              size_t ws_size, hipStream_t stream) {
  (void)n_in;
  (void)out_size;
  (void)ws_size;
  const float* hx = (const float*)d_in[0];      // [B,N,W]
  const float* hy = (const float*)d_in[1];      // [B,N,H]
  const float* labels = (const float*)d_in[2];  // [B,N,2]

  const int BN = in_sizes[2] / 2;
  const int W = in_sizes[0] / BN;
  const int H = in_sizes[1] / BN;
  const int N = N_FRAMES;  // frames per batch (reference shape)
  const int B = BN / N;

  float* coords = (float*)d_ws;               // BN*2 floats
  float* partials = coords + (size_t)BN * 2;  // B*4 floats

  row_argmax_kernel<<<BN, 32, 0, stream>>>(hx, coords, W, 0);
  row_argmax_kernel<<<BN, 32, 0, stream>>>(hy, coords, H, 1);
  physics_loss_kernel<<<B, N, 0, stream>>>(labels, coords, partials, N);
  finalize_kernel<<<1, 1, 0, stream>>>(partials, (float*)d_out, B);
}